// GCN_62629213110804
// MI455X (gfx1250) — compile-verified
//
#include <hip/hip_runtime.h>

// ---------------------------------------------------------------------------
// GCNConv (N=100000, D=30, E=3.2M) + 4-layer MLP, fused pipeline for gfx1250.
// Dense 30x30 transform runs on v_wmma_f32_16x16x4_f32 (f32 precision kept).
// Edge aggregation = streaming + L2-resident f32 atomics (the real bottleneck).
// ---------------------------------------------------------------------------

typedef __attribute__((ext_vector_type(2))) float v2f;
typedef __attribute__((ext_vector_type(8))) float v8f;

#define D_IN 30
#define DP   32   // padded feature stride

// ---------- init: agg = 0, deg = 1.0 (self-loop weight) ----------
__global__ void k_init(float* __restrict__ agg, float* __restrict__ deg, int N) {
    int stride = gridDim.x * blockDim.x;
    int i = blockIdx.x * blockDim.x + threadIdx.x;
    int total = N * DP;
    for (int idx = i; idx < total; idx += stride) agg[idx] = 0.0f;
    for (int idx = i; idx < N; idx += stride) deg[idx] = 1.0f;
}

// ---------- x[N,32] = h[N,30] @ W[30,30], zero-padded, via f32 WMMA ----------
__global__ __launch_bounds__(256) void k_conv_wmma(
    const float* __restrict__ h, const float* __restrict__ W,
    float* __restrict__ x, int N)
{
    __shared__ float sW[DP * DP];       // 32x32 zero-padded weights
    __shared__ float sA[8 * 16 * DP];   // 8 waves x (16 rows x 32 cols) A tiles

    const int tid = threadIdx.x;
    for (int i = tid; i < DP * DP; i += 256) {
        int r = i >> 5, c = i & 31;
        sW[i] = (r < D_IN && c < D_IN) ? W[r * D_IN + c] : 0.0f;
    }

    const int wave = tid >> 5;
    const int lane = tid & 31;
    const int tile = blockIdx.x * 8 + wave;
    const int m0   = tile * 16;

    float* aT = &sA[wave * 16 * DP];
    if (m0 < N) {
        for (int i = lane; i < 16 * DP; i += 32) {
            int r = i >> 5, c = i & 31;
            int gr = m0 + r;
            aT[i] = (gr < N && c < D_IN) ? h[gr * D_IN + c] : 0.0f;
        }
    }
    __syncthreads();
    if (m0 >= N) return;               // wave-uniform exit; EXEC stays all-1s

    const int half  = lane >> 4;       // 0: K pair {0,1}; 1: K pair {2,3}
    const int nlo   = lane & 15;       // A row / B,D column within 16
    const int khalf = half * 2;

    #pragma unroll
    for (int nb = 0; nb < 2; ++nb) {   // two 16-wide column blocks of D
        const int ncol = nb * 16 + nlo;
        v8f acc = {};
        #pragma unroll
        for (int kk = 0; kk < 8; ++kk) {   // K = 32 in steps of 4
            const int k0 = kk * 4 + khalf;
            v2f a, b;
            a.x = aT[nlo * DP + k0];       // A 16x4: lanes hold M=nlo, K split by half
            a.y = aT[nlo * DP + k0 + 1];
            b.x = sW[k0 * DP + ncol];      // B 4x16: lanes hold N=nlo, K split by half
            b.y = sW[(k0 + 1) * DP + ncol];
            acc = __builtin_amdgcn_wmma_f32_16x16x4_f32(
                false, a, false, b, (short)0, acc, false, false);
        }
        #pragma unroll
        for (int v = 0; v < 8; ++v) {      // D 16x16: VGPR v -> M = v + 8*half
            const int mr = m0 + v + half * 8;
            if (mr < N) x[mr * DP + ncol] = acc[v];
        }
    }
}

// ---------- deg[col] += ew (streaming scatter) ----------
__global__ void k_deg(const long long* __restrict__ col,
                      const float* __restrict__ ew,
                      float* __restrict__ deg, int E)
{
    int stride = gridDim.x * blockDim.x;
    for (int e = blockIdx.x * blockDim.x + threadIdx.x; e < E; e += stride) {
        if (e + stride < E) {
            __builtin_prefetch(&col[e + stride], 0, 0);  // global_prefetch_b8
            __builtin_prefetch(&ew[e + stride], 0, 0);
        }
        atomicAdd(&deg[(int)col[e]], ew[e]);
    }
}

// ---------- dis = rsqrt(deg) in place ----------
__global__ void k_rsqrt(float* __restrict__ deg, int N) {
    int i = blockIdx.x * blockDim.x + threadIdx.x;
    if (i < N) {
        float d = deg[i];
        deg[i] = (d > 0.0f) ? rsqrtf(d) : 0.0f;
    }
}

// ---------- agg[col] += x[row] * (dis[row]*ew*dis[col]); 8 lanes per edge ----------
__global__ __launch_bounds__(256) void k_scatter(
    const long long* __restrict__ row, const long long* __restrict__ colp,
    const float* __restrict__ ew, const float* __restrict__ dis,
    const float* __restrict__ x, float* __restrict__ agg, int E)
{
    const int gid    = blockIdx.x * blockDim.x + threadIdx.x;
    const int sub    = gid & 7;                       // 4 dims per lane
    const int stride = (gridDim.x * blockDim.x) >> 3;
    for (int e = gid >> 3; e < E; e += stride) {
        if (e + stride < E) {
            __builtin_prefetch(&row[e + stride], 0, 0);
            __builtin_prefetch(&colp[e + stride], 0, 0);
        }
        const int r = (int)row[e];
        const int c = (int)colp[e];
        const float nrm = dis[r] * ew[e] * dis[c];
        const float4 v = *reinterpret_cast<const float4*>(&x[r * DP + sub * 4]);
        float* dst = &agg[c * DP + sub * 4];
        atomicAdd(dst + 0, v.x * nrm);
        atomicAdd(dst + 1, v.y * nrm);
        atomicAdd(dst + 2, v.z * nrm);   // dims 30,31 carry zeros (padding)
        atomicAdd(dst + 3, v.w * nrm);
    }
}

// ---------- finalize: self-loop + bias + ReLU, then 30->10->10->10->1 MLP ----------
__global__ __launch_bounds__(256) void k_mlp(
    const float* __restrict__ agg, const float* __restrict__ x,
    const float* __restrict__ dis, const float* __restrict__ convb,
    const float* __restrict__ w1, const float* __restrict__ b1,
    const float* __restrict__ w2, const float* __restrict__ b2,
    const float* __restrict__ w3, const float* __restrict__ b3,
    const float* __restrict__ w4, const float* __restrict__ b4,
    float* __restrict__ out, int N)
{
    __shared__ float sw1[300], sw2[100], sw3[100], sw4[10];
    __shared__ float sb1[10], sb2[10], sb3[10], scb[D_IN], sb4;
    const int tid = threadIdx.x;
    for (int i = tid; i < 300; i += 256) sw1[i] = w1[i];
    if (tid < 100) { sw2[tid] = w2[tid]; sw3[tid] = w3[tid]; }
    if (tid < 10)  { sb1[tid] = b1[tid]; sb2[tid] = b2[tid];
                     sb3[tid] = b3[tid]; sw4[tid] = w4[tid]; }
    if (tid < D_IN) scb[tid] = convb[tid];
    if (tid == 0)   sb4 = b4[0];
    __syncthreads();

    const int i = blockIdx.x * blockDim.x + tid;
    if (i >= N) return;

    const float di = dis[i];
    const float selfn = di * di;             // self-loop: dis[i] * 1 * dis[i]

    float t0[D_IN];
    #pragma unroll
    for (int d = 0; d < D_IN; ++d) {
        float v = agg[i * DP + d] + x[i * DP + d] * selfn + scb[d];
        t0[d] = v > 0.0f ? v : 0.0f;
    }
    float t1[10];
    #pragma unroll
    for (int j = 0; j < 10; ++j) {
        float s = sb1[j];
        #pragma unroll
        for (int d = 0; d < D_IN; ++d) s += t0[d] * sw1[d * 10 + j];
        t1[j] = s > 0.0f ? s : 0.0f;
    }
    float t2[10];
    #pragma unroll
    for (int j = 0; j < 10; ++j) {
        float s = sb2[j];
        #pragma unroll
        for (int d = 0; d < 10; ++d) s += t1[d] * sw2[d * 10 + j];
        t2[j] = s > 0.0f ? s : 0.0f;
    }
    float t3[10];
    #pragma unroll
    for (int j = 0; j < 10; ++j) {
        float s = sb3[j];
        #pragma unroll
        for (int d = 0; d < 10; ++d) s += t2[d] * sw3[d * 10 + j];
        t3[j] = s > 0.0f ? s : 0.0f;
    }
    float s = sb4;
    #pragma unroll
    for (int j = 0; j < 10; ++j) s += t3[j] * sw4[j];
    out[i] = 1.0f / (1.0f + __expf(-s));
}

extern "C" void kernel_launch(void* const* d_in, const int* in_sizes, int n_in,
                              void* d_out, int out_size, void* d_ws, size_t ws_size,
                              hipStream_t stream) {
    (void)n_in; (void)out_size; (void)ws_size;
    const float*     h     = (const float*)d_in[0];
    const long long* eidx  = (const long long*)d_in[1];   // int64 [2, E]
    const float*     ew    = (const float*)d_in[2];
    const float*     convW = (const float*)d_in[3];
    const float*     convb = (const float*)d_in[4];
    const float*     w1 = (const float*)d_in[5];
    const float*     b1 = (const float*)d_in[6];
    const float*     w2 = (const float*)d_in[7];
    const float*     b2 = (const float*)d_in[8];
    const float*     w3 = (const float*)d_in[9];
    const float*     b3 = (const float*)d_in[10];
    const float*     w4 = (const float*)d_in[11];
    const float*     b4 = (const float*)d_in[12];

    const int N = in_sizes[0] / D_IN;     // 100000
    const int E = in_sizes[2];            // 3200000
    const long long* row = eidx;          // edge_index[0] = sources
    const long long* col = eidx + E;      // edge_index[1] = targets

    float* ws  = (float*)d_ws;
    float* x   = ws;                                  // [N, 32]
    float* agg = ws + (size_t)N * DP;                 // [N, 32]
    float* deg = ws + (size_t)2 * N * DP;             // [N] (becomes dis)

    k_init<<<1024, 256, 0, stream>>>(agg, deg, N);

    const int tiles = (N + 15) / 16;
    k_conv_wmma<<<(tiles + 7) / 8, 256, 0, stream>>>(h, convW, x, N);

    k_deg<<<4096, 256, 0, stream>>>(col, ew, deg, E);
    k_rsqrt<<<(N + 255) / 256, 256, 0, stream>>>(deg, N);

    const long long sthreads = (long long)E * 8;
    k_scatter<<<(int)((sthreads + 255) / 256), 256, 0, stream>>>(
        row, col, ew, deg, x, agg, E);

    k_mlp<<<(N + 255) / 256, 256, 0, stream>>>(
        agg, x, deg, convb, w1, b1, w2, b2, w3, b3, w4, b4, (float*)d_out, N);
}